// MGCNModel_28157805592958
// MI455X (gfx1250) — compile-verified
//
#include <hip/hip_runtime.h>
#include <hip/hip_bf16.h>
#include <math.h>

// ---------------------------------------------------------------------------
// MGCN forward for MI455X (gfx1250, wave32).
// Memory-bound SpMM (~1.5-2 GB irregular traffic -> ~80us @ 23.3 TB/s) +
// fp32 WMMA (V_WMMA_F32_16X16X4_F32) for all dense GEMMs.
// ---------------------------------------------------------------------------

typedef float v2f __attribute__((ext_vector_type(2)));
typedef float v8f __attribute__((ext_vector_type(8)));

#define K_DIM 64
#define U_N   30000
#define I_N   20000
#define NU_N  50000
#define LDS_STRIDE 68   // bank = (4*row + col) % 64 -> conflict-free float2 reads

__device__ __forceinline__ void atom_add_f32(float* p, float v) {
    unsafeAtomicAdd(p, v);   // hardware global_atomic_add_f32
}

// ---------------------------------------------------------------------------
// GEMM: out[M x 64] = act(A[M x Ka] @ W[Ka x 64] + bias), M % 16 == 0.
// Block = 128 threads = 4 waves; block computes 16 rows x 64 cols.
// Wave w computes the 16x16 tile at columns n0 = 16*w via fp32 WMMA,
// K-loop chunked by 64 through LDS (A row-major, W transposed).
// mode: 0 = identity, 1 = sigmoid, 2 = sigmoid * mul[row,col] (modality gate)
// ---------------------------------------------------------------------------
__global__ __launch_bounds__(128)
void gemm16x64_wmma(const float* __restrict__ A, const float* __restrict__ W,
                    const float* __restrict__ bias, const float* __restrict__ mul,
                    float* __restrict__ out, int Ka, int mode)
{
    __shared__ __align__(16) float As[16 * LDS_STRIDE];
    __shared__ __align__(16) float Bs[64 * LDS_STRIDE];   // Bs[col][k]

    const int m0   = blockIdx.x * 16;
    const int tid  = threadIdx.x;
    const int lane = tid & 31;
    const int wave = tid >> 5;
    const int hh   = lane >> 4;     // half-wave selects K-pair
    const int l16  = lane & 15;
    const int n0   = wave * 16;

    v8f acc = {};

    for (int kb = 0; kb < Ka; kb += 64) {
        // stage A tile: 16 rows x 64 k  (256 float4 loads by 128 threads)
        for (int i = tid; i < 256; i += 128) {
            const int r = i >> 4, c4 = i & 15;
            const float* src = A + (size_t)(m0 + r) * Ka + kb + c4 * 4;
            *(float4*)(&As[r * LDS_STRIDE + c4 * 4]) = *(const float4*)src;
            if (kb + 64 < Ka)
                __builtin_prefetch(src + 64, 0, 1);
        }
        // stage W chunk transposed: Bs[col][k] for k in [kb, kb+64)
        for (int i = tid; i < 4096; i += 128) {
            const int r = i >> 6, c = i & 63;
            Bs[c * LDS_STRIDE + r] = W[(size_t)(kb + r) * K_DIM + c];
        }
        __syncthreads();

        #pragma unroll
        for (int kk = 0; kk < 64; kk += 4) {
            // A frag: lane l16 = row M, half picks K = {2h, 2h+1}
            v2f a = *(const v2f*)(&As[l16 * LDS_STRIDE + kk + hh * 2]);
            // B frag: lane l16 = col N, half picks K = {2h, 2h+1}
            v2f b = *(const v2f*)(&Bs[(n0 + l16) * LDS_STRIDE + kk + hh * 2]);
            acc = __builtin_amdgcn_wmma_f32_16x16x4_f32(
                      false, a, false, b, (short)0, acc, false, false);
        }
        __syncthreads();
    }

    // epilogue: C layout -> row = m0 + 8*half + d, col = n0 + l16
    const int cn    = n0 + l16;
    const int rbase = m0 + hh * 8;
    const float bv  = bias[cn];
    #pragma unroll
    for (int d = 0; d < 8; ++d) {
        float v = acc[d] + bv;
        if (mode >= 1) v = 1.0f / (1.0f + __expf(-v));
        if (mode == 2) v *= mul[(size_t)(rbase + d) * K_DIM + cn];
        out[(size_t)(rbase + d) * K_DIM + cn] = v;
    }
}

// ---------------------------------------------------------------------------
// Attention score: att[row] += sum_n tanh( (mm @ Wq1 + bq1)[row,n] ) * Wq2[n]
// Same WMMA tiling (Ka = 64 fixed); fused tanh + dot + shfl_xor reduction.
// att must be pre-zeroed.
// ---------------------------------------------------------------------------
__global__ __launch_bounds__(128)
void gemm_att_wmma(const float* __restrict__ mm, const float* __restrict__ Wq1,
                   const float* __restrict__ bq1, const float* __restrict__ Wq2,
                   float* __restrict__ att)
{
    __shared__ __align__(16) float As[16 * LDS_STRIDE];
    __shared__ __align__(16) float Bs[64 * LDS_STRIDE];

    const int m0   = blockIdx.x * 16;
    const int tid  = threadIdx.x;
    const int lane = tid & 31;
    const int wave = tid >> 5;
    const int hh   = lane >> 4;
    const int l16  = lane & 15;
    const int n0   = wave * 16;

    for (int i = tid; i < 256; i += 128) {
        const int r = i >> 4, c4 = i & 15;
        *(float4*)(&As[r * LDS_STRIDE + c4 * 4]) =
            *(const float4*)(mm + (size_t)(m0 + r) * K_DIM + c4 * 4);
    }
    for (int i = tid; i < 4096; i += 128) {
        const int r = i >> 6, c = i & 63;
        Bs[c * LDS_STRIDE + r] = Wq1[(size_t)r * K_DIM + c];
    }
    __syncthreads();

    v8f acc = {};
    #pragma unroll
    for (int kk = 0; kk < 64; kk += 4) {
        v2f a = *(const v2f*)(&As[l16 * LDS_STRIDE + kk + hh * 2]);
        v2f b = *(const v2f*)(&Bs[(n0 + l16) * LDS_STRIDE + kk + hh * 2]);
        acc = __builtin_amdgcn_wmma_f32_16x16x4_f32(
                  false, a, false, b, (short)0, acc, false, false);
    }

    const int cn = n0 + l16;
    const float w2 = Wq2[cn];
    const float bv = bq1[cn];
    #pragma unroll
    for (int d = 0; d < 8; ++d) {
        float v = tanhf(acc[d] + bv) * w2;
        // reduce across the 16 lanes of this half-wave (same rows)
        v += __shfl_xor(v, 1, 32);
        v += __shfl_xor(v, 2, 32);
        v += __shfl_xor(v, 4, 32);
        v += __shfl_xor(v, 8, 32);
        if (l16 == 0)
            atom_add_f32(&att[m0 + hh * 8 + d], v);
    }
}

// ---------------------------------------------------------------------------
// SpMM with atomics: y[rows[e], :] += vals[e] * x[cols[e], :], K = 64.
// One wave per edge: 2 coalesced floats per lane gather, 2 fp32 atomics.
// y must be pre-zeroed over the touched row range.
// ---------------------------------------------------------------------------
__global__ __launch_bounds__(256)
void spmm_atomic(const int* __restrict__ rows, const int* __restrict__ cols,
                 const float* __restrict__ vals, const float* __restrict__ x,
                 float* __restrict__ y, int nE)
{
    const int e = blockIdx.x * 8 + (threadIdx.x >> 5);
    if (e >= nE) return;
    const int lane = threadIdx.x & 31;
    const int r = rows[e];
    const int c = cols[e];
    const float v = vals[e];
    const float* xs = x + (size_t)c * K_DIM;
    float* yd = y + (size_t)r * K_DIM;
    atom_add_f32(yd + lane,      v * xs[lane]);
    atom_add_f32(yd + lane + 32, v * xs[lane + 32]);
}

// ----------------------------- elementwise ---------------------------------
__global__ void concat_uv(const float* __restrict__ Gu, const float* __restrict__ Gi,
                          float* __restrict__ ego)
{
    const size_t i = (size_t)blockIdx.x * 256 + threadIdx.x;
    const size_t UK = (size_t)U_N * K_DIM;
    const size_t N  = (size_t)NU_N * K_DIM;
    if (i >= N) return;
    ego[i] = (i < UK) ? Gu[i] : Gi[i - UK];
}

__global__ void mean3_inplace(float* __restrict__ c, const float* __restrict__ h1,
                              const float* __restrict__ h2, size_t n)
{
    const size_t i = (size_t)blockIdx.x * 256 + threadIdx.x;
    if (i >= n) return;
    c[i] = (c[i] + h1[i] + h2[i]) * (1.0f / 3.0f);
}

__global__ void copy_f32(const float* __restrict__ src, float* __restrict__ dst, size_t n)
{
    const size_t i = (size_t)blockIdx.x * 256 + threadIdx.x;
    if (i >= n) return;
    dst[i] = src[i];
}

__global__ void softmax2(const float* __restrict__ av, const float* __restrict__ at,
                         float* __restrict__ w0, int n)
{
    const int i = blockIdx.x * 256 + threadIdx.x;
    if (i >= n) return;
    const float a = av[i], b = at[i];
    const float m = fmaxf(a, b);
    const float ea = __expf(a - m), eb = __expf(b - m);
    w0[i] = ea / (ea + eb);
}

// dvec[0:64]  = (mm_v - common) row 0 ; dvec[64:128] = (mm_t - common) row 1
__global__ void row_delta(const float* __restrict__ mm_v, const float* __restrict__ mm_t,
                          const float* __restrict__ w0, float* __restrict__ dvec)
{
    const int k = threadIdx.x;   // 64 threads
    const float w00 = w0[0], w01 = w0[1];
    const float c0 = w00 * mm_v[k] + (1.0f - w00) * mm_t[k];
    dvec[k] = mm_v[k] - c0;
    const float c1 = w01 * mm_v[K_DIM + k] + (1.0f - w01) * mm_t[K_DIM + k];
    dvec[K_DIM + k] = mm_t[K_DIM + k] - c1;
}

__global__ void final_combine(const float* __restrict__ content,
                              const float* __restrict__ mm_v, const float* __restrict__ mm_t,
                              const float* __restrict__ pref_v, const float* __restrict__ pref_t,
                              const float* __restrict__ w0, const float* __restrict__ dvec,
                              float* __restrict__ out, size_t n)
{
    const size_t i = (size_t)blockIdx.x * 256 + threadIdx.x;
    if (i >= n) return;
    const size_t row = i >> 6;
    const int k = (int)(i & 63);
    const float w = w0[row];
    const float common = w * mm_v[i] + (1.0f - w) * mm_t[i];
    const float side = pref_v[i] * dvec[k] + pref_t[i] * dvec[K_DIM + k];
    out[i] = content[i] + (side + common) * (1.0f / 3.0f);
}

// ---------------------------------------------------------------------------
extern "C" void kernel_launch(void* const* d_in, const int* in_sizes, int n_in,
                              void* d_out, int out_size, void* d_ws, size_t ws_size,
                              hipStream_t stream)
{
    (void)n_in; (void)out_size; (void)ws_size;

    const float* Gu      = (const float*)d_in[0];
    const float* Gi      = (const float*)d_in[1];
    const float* feat_v  = (const float*)d_in[2];
    const float* feat_t  = (const float*)d_in[3];
    const float* Wp_v    = (const float*)d_in[4];
    const float* bp_v    = (const float*)d_in[5];
    const float* Wp_t    = (const float*)d_in[6];
    const float* bp_t    = (const float*)d_in[7];
    const float* Wg_v    = (const float*)d_in[8];
    const float* bg_v    = (const float*)d_in[9];
    const float* Wg_t    = (const float*)d_in[10];
    const float* bg_t    = (const float*)d_in[11];
    const float* Wpref_v = (const float*)d_in[12];
    const float* bpref_v = (const float*)d_in[13];
    const float* Wpref_t = (const float*)d_in[14];
    const float* bpref_t = (const float*)d_in[15];
    const float* Wq1     = (const float*)d_in[16];
    const float* bq1     = (const float*)d_in[17];
    const float* Wq2     = (const float*)d_in[18];
    const float* adj_vals   = (const float*)d_in[19];
    const float* sim_vals_v = (const float*)d_in[20];
    const float* sim_vals_t = (const float*)d_in[21];
    const float* r_vals     = (const float*)d_in[22];
    const int* adj_rows   = (const int*)d_in[23];
    const int* adj_cols   = (const int*)d_in[24];
    const int* sim_rows_v = (const int*)d_in[25];
    const int* sim_cols_v = (const int*)d_in[26];
    const int* sim_rows_t = (const int*)d_in[27];
    const int* sim_cols_t = (const int*)d_in[28];
    const int* r_rows     = (const int*)d_in[29];
    const int* r_cols     = (const int*)d_in[30];

    const int nAdj = in_sizes[23];   // 2,000,000
    const int nSim = in_sizes[25];   //   200,000
    const int nR   = in_sizes[29];   // 1,000,000

    // ---- workspace layout (fp32), with liveness-based reuse ----
    float* ws = (float*)d_ws;
    const size_t IK  = (size_t)I_N * K_DIM;    // 1.28M floats
    const size_t NUK = (size_t)NU_N * K_DIM;   // 3.20M floats
    const size_t UK  = (size_t)U_N * K_DIM;

    float* proj_v  = ws;               // IK   (dead after gating -> itemv)
    float* proj_t  = ws + IK;          // IK   (dead after gating -> itemt)
    float* gated_v = ws + 2 * IK;      // IK
    float* gated_t = ws + 3 * IK;      // IK
    float* content = ws + 4 * IK;      // NUK  (ego -> content)
    float* h1      = content + NUK;    // NUK  (dead after mean3 -> pref_v)
    float* h2      = h1 + NUK;         // NUK  (dead after mean3 -> pref_t)
    float* mm_v    = h2 + NUK;         // NUK
    float* mm_t    = mm_v + NUK;       // NUK
    float* att_v   = mm_t + NUK;       // NU
    float* att_t   = att_v + NU_N;     // NU
    float* w0      = att_t + NU_N;     // NU
    float* dvec    = w0 + NU_N;        // 128

    const dim3 b128(128), b256(256);

    // 1) modality projections:  proj = feat @ Wp + bp
    gemm16x64_wmma<<<I_N / 16, b128, 0, stream>>>(feat_v, Wp_v, bp_v, nullptr, proj_v, 2048, 0);
    gemm16x64_wmma<<<I_N / 16, b128, 0, stream>>>(feat_t, Wp_t, bp_t, nullptr, proj_t, 768, 0);

    // 2) gating:  gated = Gi * sigmoid(proj @ Wg + bg)
    gemm16x64_wmma<<<I_N / 16, b128, 0, stream>>>(proj_v, Wg_v, bg_v, Gi, gated_v, 64, 2);
    gemm16x64_wmma<<<I_N / 16, b128, 0, stream>>>(proj_t, Wg_t, bg_t, Gi, gated_t, 64, 2);

    // 3) LightGCN hops over UI graph
    concat_uv<<<(unsigned)((NUK + 255) / 256), b256, 0, stream>>>(Gu, Gi, content);
    hipMemsetAsync(h1, 0, NUK * sizeof(float), stream);
    spmm_atomic<<<(nAdj + 7) / 8, b256, 0, stream>>>(adj_rows, adj_cols, adj_vals, content, h1, nAdj);
    hipMemsetAsync(h2, 0, NUK * sizeof(float), stream);
    spmm_atomic<<<(nAdj + 7) / 8, b256, 0, stream>>>(adj_rows, adj_cols, adj_vals, h1, h2, nAdj);
    mean3_inplace<<<(unsigned)((NUK + 255) / 256), b256, 0, stream>>>(content, h1, h2, NUK);

    // 4) item-item kNN hop per modality (reuse proj buffers)
    float* itemv = proj_v;
    float* itemt = proj_t;
    hipMemsetAsync(itemv, 0, IK * sizeof(float), stream);
    spmm_atomic<<<(nSim + 7) / 8, b256, 0, stream>>>(sim_rows_v, sim_cols_v, sim_vals_v, gated_v, itemv, nSim);
    hipMemsetAsync(itemt, 0, IK * sizeof(float), stream);
    spmm_atomic<<<(nSim + 7) / 8, b256, 0, stream>>>(sim_rows_t, sim_cols_t, sim_vals_t, gated_t, itemt, nSim);

    // 5) lift to users via R, concat -> mm
    hipMemsetAsync(mm_v, 0, UK * sizeof(float), stream);
    spmm_atomic<<<(nR + 7) / 8, b256, 0, stream>>>(r_rows, r_cols, r_vals, itemv, mm_v, nR);
    copy_f32<<<(unsigned)((IK + 255) / 256), b256, 0, stream>>>(itemv, mm_v + UK, IK);
    hipMemsetAsync(mm_t, 0, UK * sizeof(float), stream);
    spmm_atomic<<<(nR + 7) / 8, b256, 0, stream>>>(r_rows, r_cols, r_vals, itemt, mm_t, nR);
    copy_f32<<<(unsigned)((IK + 255) / 256), b256, 0, stream>>>(itemt, mm_t + UK, IK);

    // 6) attention scores (fused tanh + dot + reduce)
    hipMemsetAsync(att_v, 0, NU_N * sizeof(float), stream);
    hipMemsetAsync(att_t, 0, NU_N * sizeof(float), stream);
    gemm_att_wmma<<<NU_N / 16, b128, 0, stream>>>(mm_v, Wq1, bq1, Wq2, att_v);
    gemm_att_wmma<<<NU_N / 16, b128, 0, stream>>>(mm_t, Wq1, bq1, Wq2, att_t);

    // 7) modality preferences (reuse h1/h2)
    float* pref_v = h1;
    float* pref_t = h2;
    gemm16x64_wmma<<<NU_N / 16, b128, 0, stream>>>(content, Wpref_v, bpref_v, nullptr, pref_v, 64, 1);
    gemm16x64_wmma<<<NU_N / 16, b128, 0, stream>>>(content, Wpref_t, bpref_t, nullptr, pref_t, 64, 1);

    // 8) softmax over 2 modalities, broadcast rows, final combine
    softmax2<<<(NU_N + 255) / 256, b256, 0, stream>>>(att_v, att_t, w0, NU_N);
    row_delta<<<1, 64, 0, stream>>>(mm_v, mm_t, w0, dvec);
    final_combine<<<(unsigned)((NUK + 255) / 256), b256, 0, stream>>>(
        content, mm_v, mm_t, pref_v, pref_t, w0, dvec, (float*)d_out, NUK);
}